// GroupContrastiveLoss_51427938402771
// MI455X (gfx1250) — compile-verified
//
#include <hip/hip_runtime.h>

#define KCL   16
#define BETA  0.3f

typedef __attribute__((ext_vector_type(2))) float v2f;
typedef __attribute__((ext_vector_type(8))) float v8f;

// meta (ints): [0,16) counts | [16,32) cursors | [32,48) segStart |
//              [48,64) tilesPerDim | [64,81) tilePrefix | [81] totalTiles
#define M_COUNT  0
#define M_CURSOR 16
#define M_SEG    32
#define M_TD     48
#define M_TP     64
#define M_TOTAL  81

__global__ void gcl_init(int* meta, float* accum, int* slabel, int npad) {
  int i = blockIdx.x * blockDim.x + threadIdx.x;
  if (i < npad) slabel[i] = -1;        // pad slots stay -1 -> masked out
  if (i < 32)  meta[i] = 0;            // counts + cursors
  if (i < 64)  accum[i] = 0.f;         // per-cluster {pos_sum,pos_cnt,neg_sum,neg_cnt}
}

__global__ void gcl_count(const int* __restrict__ clusters, int* meta, int n) {
  int i = blockIdx.x * blockDim.x + threadIdx.x;
  if (i < n) atomicAdd(&meta[M_COUNT + (clusters[i] & (KCL - 1))], 1);
}

__global__ void gcl_plan(int* meta) {
  if (threadIdx.x == 0 && blockIdx.x == 0) {
    int s = 0, tp = 0;
    for (int c = 0; c < KCL; ++c) {
      int cnt = meta[M_COUNT + c];
      int pad = (cnt + 63) & ~63;      // segment padded to 64-row macro tiles
      int td  = pad >> 6;
      meta[M_SEG + c] = s;
      meta[M_TD  + c] = td;
      meta[M_TP  + c] = tp;
      tp += td * td;
      s  += pad;
    }
    meta[M_TP + KCL] = tp;
    meta[M_TOTAL]    = tp;
  }
}

__global__ void gcl_scatter(const int* __restrict__ clusters,
                            const int* __restrict__ labels,
                            int* meta, int* slabel, int* perm, int n) {
  int i = blockIdx.x * blockDim.x + threadIdx.x;
  if (i < n) {
    int c    = clusters[i] & (KCL - 1);
    int slot = meta[M_SEG + c] + atomicAdd(&meta[M_CURSOR + c], 1);
    perm[i]      = slot;
    slabel[slot] = labels[i];
  }
}

// Normalize row i and write it into the WMMA-fragment-native panel layout:
// panel of 16 rows; slab per k4 = 64 floats: lanes 0..15 -> (row, 4k4+{0,1}),
// lanes 16..31 -> (row, 4k4+{2,3}).  One thread handles one k4 slab.
__global__ void gcl_normalize(const float* __restrict__ feat,
                              const int* __restrict__ perm,
                              float* __restrict__ panel,
                              int d, int panelStride) {
  __shared__ float red[256];
  const int row = blockIdx.x;
  const int t   = threadIdx.x;                       // k4 index, d/4 threads
  const float4 v = ((const float4*)(feat + (size_t)row * d))[t];
  red[t] = v.x * v.x + v.y * v.y + v.z * v.z + v.w * v.w;
  __syncthreads();
  for (int off = blockDim.x >> 1; off > 0; off >>= 1) {
    if (t < off) red[t] += red[t + off];
    __syncthreads();
  }
  const float scale = 1.0f / fmaxf(sqrtf(red[0]), 1e-12f);
  const int slot = perm[row];
  float* base = panel + (size_t)(slot >> 4) * panelStride
                      + (size_t)t * 64 + (slot & 15) * 2;
  base[0]  = v.x * scale;   // k4*4+0  -> lane (slot&15)      elem 0
  base[1]  = v.y * scale;   // k4*4+1  -> lane (slot&15)      elem 1
  base[32] = v.z * scale;   // k4*4+2  -> lane (slot&15)+16   elem 0
  base[33] = v.w * scale;   // k4*4+3  -> lane (slot&15)+16   elem 1
}

// Persistent blocks: 128 threads = 4 waves, workgroup owns a 64x64 intra-cluster
// sim macro-tile; each wave computes 2x2 f32 WMMA tiles over K = d.
__global__ void gcl_gemm(const float* __restrict__ panel,
                         const int* __restrict__ slabel,
                         const int* __restrict__ meta,
                         float* __restrict__ accum,
                         int d, int panelStride) {
  const int tid  = threadIdx.x;
  const int lane = tid & 31;
  const int wv   = tid >> 5;                 // 0..3
  const int wvI  = (wv >> 1) << 5;           // 0 or 32 (rows)
  const int wvJ  = (wv & 1) << 5;            // 0 or 32 (cols)
  const int k4n  = d >> 2;
  const int total = meta[M_TOTAL];
  const int nsub = lane & 15;                // C/D: n = lane%16
  const int mAdd = (lane >> 4) << 3;         // C/D: m = r + 8*(lane>=16)

  for (int w = blockIdx.x; w < total; w += gridDim.x) {
    int c = 0;                                         // uniform decode
    while (c < KCL - 1 && w >= meta[M_TP + c + 1]) ++c;
    const int local = w - meta[M_TP + c];
    const int td    = meta[M_TD + c];
    const int ti    = local / td;
    const int tj    = local - ti * td;
    const int seg   = meta[M_SEG + c];
    const int I0    = seg + ti * 64 + wvI;
    const int J0    = seg + tj * 64 + wvJ;

    const float* pa0 = panel + (size_t)(I0 >> 4) * panelStride + 2 * lane;
    const float* pa1 = pa0 + panelStride;
    const float* pb0 = panel + (size_t)(J0 >> 4) * panelStride + 2 * lane;
    const float* pb1 = pb0 + panelStride;

    v8f a00 = {}, a01 = {}, a10 = {}, a11 = {};
#pragma unroll 4
    for (int k4 = 0; k4 < k4n; ++k4) {
      const size_t o = (size_t)k4 * 64;
      v2f fa0 = *(const v2f*)(pa0 + o);      // A frag, tile row 0
      v2f fa1 = *(const v2f*)(pa1 + o);      // A frag, tile row 1
      v2f fb0 = *(const v2f*)(pb0 + o);      // B frag = same striping, col panel
      v2f fb1 = *(const v2f*)(pb1 + o);
      a00 = __builtin_amdgcn_wmma_f32_16x16x4_f32(false, fa0, false, fb0, (short)0, a00, false, false);
      a01 = __builtin_amdgcn_wmma_f32_16x16x4_f32(false, fa0, false, fb1, (short)0, a01, false, false);
      a10 = __builtin_amdgcn_wmma_f32_16x16x4_f32(false, fa1, false, fb0, (short)0, a10, false, false);
      a11 = __builtin_amdgcn_wmma_f32_16x16x4_f32(false, fa1, false, fb1, (short)0, a11, false, false);
    }

    float ps = 0.f, pc = 0.f, ns = 0.f, nc = 0.f;
#pragma unroll
    for (int ta = 0; ta < 2; ++ta) {
#pragma unroll
      for (int tb = 0; tb < 2; ++tb) {
        v8f acc = (ta == 0) ? ((tb == 0) ? a00 : a01)
                            : ((tb == 0) ? a10 : a11);
        const int I  = I0 + ta * 16;
        const int J  = J0 + tb * 16;
        const int jj = J + nsub;
        const int lj = slabel[jj];
#pragma unroll
        for (int r = 0; r < 8; ++r) {
          const int ii = I + mAdd + r;
          const int li = slabel[ii];
          const float s = acc[r];
          if (li >= 0 && lj >= 0 && ii != jj) {   // valid, same cluster, not diag
            if (li == lj) {
              const float dd = 1.0f - s;
              ps += dd * dd; pc += 1.0f;
            } else {
              nc += 1.0f;
              const float dd = s - BETA;
              if (dd > 0.f) ns += dd * dd;
            }
          }
        }
      }
    }
#pragma unroll
    for (int off = 16; off > 0; off >>= 1) {
      ps += __shfl_xor(ps, off, 32);
      pc += __shfl_xor(pc, off, 32);
      ns += __shfl_xor(ns, off, 32);
      nc += __shfl_xor(nc, off, 32);
    }
    if (lane == 0) {
      atomicAdd(&accum[4 * c + 0], ps);
      atomicAdd(&accum[4 * c + 1], pc);
      atomicAdd(&accum[4 * c + 2], ns);
      atomicAdd(&accum[4 * c + 3], nc);
    }
  }
}

__global__ void gcl_finalize(const int* __restrict__ meta,
                             const float* __restrict__ accum,
                             float* __restrict__ out) {
  const int lane = threadIdx.x;
  float loss = 0.f, pres = 0.f;
  if (lane < KCL) {
    if (meta[M_COUNT + lane] > 0) pres = 1.f;
    const float psum = accum[4 * lane + 0], pcnt = accum[4 * lane + 1];
    const float nsum = accum[4 * lane + 2], ncnt = accum[4 * lane + 3];
    const float pl = (pcnt > 0.f) ? psum / fmaxf(pcnt, 1.f) : 0.f;
    const float nl = (ncnt > 0.f) ? nsum / fmaxf(ncnt, 1.f) : 0.f;
    loss = pl + nl;
  }
  for (int off = 16; off > 0; off >>= 1) {
    loss += __shfl_xor(loss, off, 32);
    pres += __shfl_xor(pres, off, 32);
  }
  if (lane == 0) out[0] = loss / fmaxf(pres, 1.f);
}

extern "C" void kernel_launch(void* const* d_in, const int* in_sizes, int n_in,
                              void* d_out, int out_size, void* d_ws, size_t ws_size,
                              hipStream_t stream) {
  const float* feat     = (const float*)d_in[0];
  const int*   labels   = (const int*)d_in[1];
  const int*   clusters = (const int*)d_in[2];
  float*       out      = (float*)d_out;

  const int n    = in_sizes[1];            // B*T = 8192
  const int d    = in_sizes[0] / n;        // 1024
  const int npad = n + 64 * KCL;           // worst-case padded rows
  const int panelStride = 16 * d;          // floats per 16-row panel

  char*  ws         = (char*)d_ws;
  float* panel      = (float*)ws;
  size_t panelBytes = (size_t)npad * d * sizeof(float);
  int*   meta       = (int*)(ws + panelBytes);
  float* accum      = (float*)(meta + 128);
  int*   slabel     = (int*)(accum + 64);
  int*   perm       = slabel + npad;

  gcl_init    <<<(npad + 255) / 256, 256, 0, stream>>>(meta, accum, slabel, npad);
  gcl_count   <<<(n + 255) / 256,   256, 0, stream>>>(clusters, meta, n);
  gcl_plan    <<<1, 1, 0, stream>>>(meta);
  gcl_scatter <<<(n + 255) / 256,   256, 0, stream>>>(clusters, labels, meta, slabel, perm, n);
  gcl_normalize<<<n, d / 4, 0, stream>>>(feat, perm, panel, d, panelStride);
  gcl_gemm    <<<2048, 128, 0, stream>>>(panel, slabel, meta, accum, d, panelStride);
  gcl_finalize<<<1, 32, 0, stream>>>(meta, accum, out);
}